// ALEGridUpdate_51685636440549
// MI455X (gfx1250) — compile-verified
//
#include <hip/hip_runtime.h>
#include <hip/hip_bf16.h>

// ---------------------------------------------------------------------------
// Problem constants (match reference)
// ---------------------------------------------------------------------------
#define NN   100000      // nodes
#define NE   1600000     // edges before self loops
#define DD   128         // hidden dim
#define HH   8           // heads
#define CC   16          // channels per head
#define TOTE (NE + NN)   // edges incl. self loops

typedef __attribute__((ext_vector_type(2))) float v2f;
typedef __attribute__((ext_vector_type(8))) float v8f;

// ---------------------------------------------------------------------------
// Helpers
// ---------------------------------------------------------------------------
__device__ __forceinline__ unsigned f2ord(float f) {
  unsigned u = __float_as_uint(f);
  return u ^ ((u >> 31) ? 0xFFFFFFFFu : 0x80000000u);   // monotone total order
}
__device__ __forceinline__ float ord2f(unsigned u) {
  unsigned v = u ^ ((u >> 31) ? 0x80000000u : 0xFFFFFFFFu);
  return __uint_as_float(v);
}
__device__ __forceinline__ float lrelu(float x) { return x > 0.f ? x : 0.2f * x; }

__device__ __forceinline__ float wave_sum(float v) {
#pragma unroll
  for (int o = 16; o > 0; o >>= 1) v += __shfl_xor(v, o, 32);
  return v;
}

// ---------------------------------------------------------------------------
// Row statistics for LayerNorm. One wave (32 lanes) per row.
// ---------------------------------------------------------------------------
__global__ __launch_bounds__(256) void rowstat3_kernel(
    const float* __restrict__ a, const float* __restrict__ b,
    const float* __restrict__ c, float* __restrict__ mu,
    float* __restrict__ rstd) {
  const int wave = (blockIdx.x * blockDim.x + threadIdx.x) >> 5;
  const int lane = threadIdx.x & 31;
  if (wave >= NN) return;
  float s = 0.f, ss = 0.f;
  for (int k = lane; k < 3 * DD; k += 32) {
    const float* p = (k < DD) ? a : (k < 2 * DD ? b : c);
    float v = p[wave * DD + (k & (DD - 1))];
    s += v; ss += v * v;
  }
  s = wave_sum(s); ss = wave_sum(ss);
  if (lane == 0) {
    float m = s * (1.f / (3.f * DD));
    float var = ss * (1.f / (3.f * DD)) - m * m;
    mu[wave] = m;
    rstd[wave] = rsqrtf(var + 1e-5f);
  }
}

__global__ __launch_bounds__(256) void rowstat1_kernel(
    const float* __restrict__ x, float* __restrict__ mu,
    float* __restrict__ rstd) {
  const int wave = (blockIdx.x * blockDim.x + threadIdx.x) >> 5;
  const int lane = threadIdx.x & 31;
  if (wave >= NN) return;
  float s = 0.f, ss = 0.f;
#pragma unroll
  for (int k = lane; k < DD; k += 32) {
    float v = x[wave * DD + k];
    s += v; ss += v * v;
  }
  s = wave_sum(s); ss = wave_sum(ss);
  if (lane == 0) {
    float m = s * (1.f / DD);
    float var = ss * (1.f / DD) - m * m;
    mu[wave] = m;
    rstd[wave] = rsqrtf(var + 1e-5f);
  }
}

// ---------------------------------------------------------------------------
// WMMA f32 GEMM: out[N,128] = LN(cat(ps,pf,pb)) @ W.T (+bias).  K = 384.
// One wave computes a 16x16 output tile using V_WMMA_F32_16X16X4_F32.
// A fragment (16x4, ISA layout): lanes 0-15 carry K=k0,k0+1 of row M=lane;
// lanes 16-31 carry K=k0+2,k0+3.  B fragment (4x16) symmetric with N=lane&15.
// ---------------------------------------------------------------------------
__global__ __launch_bounds__(256) void gemm_cat_wmma(
    const float* __restrict__ ps, const float* __restrict__ pf,
    const float* __restrict__ pb, const float* __restrict__ mu,
    const float* __restrict__ rstd, const float* __restrict__ g,
    const float* __restrict__ bln, const float* __restrict__ W,
    const float* __restrict__ bias, float* __restrict__ out) {
  const int wave = (blockIdx.x * blockDim.x + threadIdx.x) >> 5;
  const int lane = threadIdx.x & 31;
  const int ct = wave & 7;            // 128/16 = 8 column tiles
  const int rt = wave >> 3;
  if (rt >= NN / 16) return;
  const int m0 = rt * 16, n0 = ct * 16;
  const int mA = m0 + (lane & 15);
  const int nB = n0 + (lane & 15);
  const int kf = (lane >> 4) << 1;    // 0 or 2
  const float muv = mu[mA], rsv = rstd[mA];
  v8f acc = {0.f, 0.f, 0.f, 0.f, 0.f, 0.f, 0.f, 0.f};
  for (int k0 = 0; k0 < 3 * DD; k0 += 4) {
    const int k1 = k0 + kf, k2 = k1 + 1;
    const float* p1 = (k1 < DD) ? ps : (k1 < 2 * DD ? pf : pb);
    const float* p2 = (k2 < DD) ? ps : (k2 < 2 * DD ? pf : pb);
    v2f A, B;
    A.x = (p1[mA * DD + (k1 & (DD - 1))] - muv) * rsv * g[k1] + bln[k1];
    A.y = (p2[mA * DD + (k2 & (DD - 1))] - muv) * rsv * g[k2] + bln[k2];
    B.x = W[nB * (3 * DD) + k1];
    B.y = W[nB * (3 * DD) + k2];
    acc = __builtin_amdgcn_wmma_f32_16x16x4_f32(false, A, false, B, (short)0,
                                                acc, false, false);
  }
  const float bv = bias ? bias[n0 + (lane & 15)] : 0.f;
  const int mr = m0 + ((lane >> 4) << 3);
#pragma unroll
  for (int r = 0; r < 8; ++r)
    out[(mr + r) * DD + n0 + (lane & 15)] = acc[r] + bv;
}

// ---------------------------------------------------------------------------
// WMMA f32 GEMM: out[N,128] = LN(x) @ W.T (+bias).  K = 128.
// ---------------------------------------------------------------------------
__global__ __launch_bounds__(256) void gemm_ln_wmma(
    const float* __restrict__ x, const float* __restrict__ mu,
    const float* __restrict__ rstd, const float* __restrict__ g,
    const float* __restrict__ bln, const float* __restrict__ W,
    const float* __restrict__ bias, float* __restrict__ out) {
  const int wave = (blockIdx.x * blockDim.x + threadIdx.x) >> 5;
  const int lane = threadIdx.x & 31;
  const int ct = wave & 7;
  const int rt = wave >> 3;
  if (rt >= NN / 16) return;
  const int m0 = rt * 16, n0 = ct * 16;
  const int mA = m0 + (lane & 15);
  const int nB = n0 + (lane & 15);
  const int kf = (lane >> 4) << 1;
  const float muv = mu[mA], rsv = rstd[mA];
  v8f acc = {0.f, 0.f, 0.f, 0.f, 0.f, 0.f, 0.f, 0.f};
#pragma unroll 4
  for (int k0 = 0; k0 < DD; k0 += 4) {
    const int k1 = k0 + kf, k2 = k1 + 1;
    v2f A, B;
    A.x = (x[mA * DD + k1] - muv) * rsv * g[k1] + bln[k1];
    A.y = (x[mA * DD + k2] - muv) * rsv * g[k2] + bln[k2];
    B.x = W[nB * DD + k1];
    B.y = W[nB * DD + k2];
    acc = __builtin_amdgcn_wmma_f32_16x16x4_f32(false, A, false, B, (short)0,
                                                acc, false, false);
  }
  const float bv = bias ? bias[n0 + (lane & 15)] : 0.f;
  const int mr = m0 + ((lane >> 4) << 3);
#pragma unroll
  for (int r = 0; r < 8; ++r)
    out[(mr + r) * DD + n0 + (lane & 15)] = acc[r] + bv;
}

// ---------------------------------------------------------------------------
// Attention coefficients: a_src[n,h] = <xh[n,h,:], att_src[h,:]>, same for dst.
// ---------------------------------------------------------------------------
__global__ __launch_bounds__(256) void attn_kernel(
    const float* __restrict__ xh, const float* __restrict__ att_src,
    const float* __restrict__ att_dst, float* __restrict__ a_src,
    float* __restrict__ a_dst) {
  const int t = blockIdx.x * blockDim.x + threadIdx.x;
  if (t >= NN * HH) return;
  const int n = t >> 3, h = t & 7;
  const float* xr = xh + n * DD + h * CC;
  const float* as = att_src + h * CC;
  const float* ad = att_dst + h * CC;
  float s1 = 0.f, s2 = 0.f;
#pragma unroll
  for (int c = 0; c < CC; ++c) {
    float v = xr[c];
    s1 += v * as[c];
    s2 += v * ad[c];
  }
  a_src[t] = s1;
  a_dst[t] = s2;
}

__global__ __launch_bounds__(256) void init_kernel(unsigned* __restrict__ amax,
                                                   float* __restrict__ denom) {
  const int t = blockIdx.x * blockDim.x + threadIdx.x;
  if (t < NN * HH) { amax[t] = 0u; denom[t] = 0.f; }
}

// ---------------------------------------------------------------------------
// Edge pass 1: segment max of leaky_relu(a_src[src]+a_dst[dst]) over dst.
// Ordered-uint atomicMax (L2 resident).
// ---------------------------------------------------------------------------
__global__ __launch_bounds__(256) void edge_max_kernel(
    const int* __restrict__ esrc, const int* __restrict__ edst,
    const float* __restrict__ a_src, const float* __restrict__ a_dst,
    unsigned* __restrict__ amax) {
  const int e = blockIdx.x * blockDim.x + threadIdx.x;
  if (e >= TOTE) return;
  int s, d;
  if (e < NE) { s = esrc[e]; d = edst[e]; } else { s = d = e - NE; }
#pragma unroll
  for (int h = 0; h < HH; ++h) {
    float al = lrelu(a_src[s * HH + h] + a_dst[d * HH + h]);
    atomicMax(&amax[d * HH + h], f2ord(al));
  }
}

// Edge pass 2: denom = segment_sum(exp(alpha - amax[dst]))
__global__ __launch_bounds__(256) void edge_sum_kernel(
    const int* __restrict__ esrc, const int* __restrict__ edst,
    const float* __restrict__ a_src, const float* __restrict__ a_dst,
    const unsigned* __restrict__ amax, float* __restrict__ denom) {
  const int e = blockIdx.x * blockDim.x + threadIdx.x;
  if (e >= TOTE) return;
  int s, d;
  if (e < NE) { s = esrc[e]; d = edst[e]; } else { s = d = e - NE; }
#pragma unroll
  for (int h = 0; h < HH; ++h) {
    float al = lrelu(a_src[s * HH + h] + a_dst[d * HH + h]);
    float ex = expf(al - ord2f(amax[d * HH + h]));
    atomicAdd(&denom[d * HH + h], ex);
  }
}

// Edge pass 3: out[dst] += w_e * xh[src].  One wave per edge, 4 channels/lane.
__global__ __launch_bounds__(256) void edge_aggr_kernel(
    const int* __restrict__ esrc, const int* __restrict__ edst,
    const float* __restrict__ a_src, const float* __restrict__ a_dst,
    const unsigned* __restrict__ amax, const float* __restrict__ denom,
    const float* __restrict__ xh, float* __restrict__ out) {
  const int wave = (blockIdx.x * blockDim.x + threadIdx.x) >> 5;
  const int lane = threadIdx.x & 31;
  if (wave >= TOTE) return;
  int s, d;
  if (wave < NE) { s = esrc[wave]; d = edst[wave]; } else { s = d = wave - NE; }
  const int h = lane >> 2;                     // 4 lanes per head
  float al = lrelu(a_src[s * HH + h] + a_dst[d * HH + h]);
  float w = expf(al - ord2f(amax[d * HH + h])) / denom[d * HH + h];
  const float4 v = *((const float4*)(xh + (size_t)s * DD) + lane);
  float* o = out + (size_t)d * DD + lane * 4;
  atomicAdd(o + 0, v.x * w);
  atomicAdd(o + 1, v.y * w);
  atomicAdd(o + 2, v.z * w);
  atomicAdd(o + 3, v.w * w);
}

// ---------------------------------------------------------------------------
// Host-side orchestration
// ---------------------------------------------------------------------------
extern "C" void kernel_launch(void* const* d_in, const int* in_sizes, int n_in,
                              void* d_out, int out_size, void* d_ws,
                              size_t ws_size, hipStream_t stream) {
  (void)in_sizes; (void)n_in; (void)out_size; (void)ws_size;

  const float* ps      = (const float*)d_in[1];
  const float* pf      = (const float*)d_in[2];
  const float* pb      = (const float*)d_in[3];
  const int*   ei      = (const int*)d_in[4];
  const float* ln0_g   = (const float*)d_in[5];
  const float* ln0_b   = (const float*)d_in[6];
  const float* fc_W    = (const float*)d_in[7];
  const float* fc_b    = (const float*)d_in[8];
  const float* lnp_g   = (const float*)d_in[9];
  const float* lnp_b   = (const float*)d_in[10];
  const float* Wp      = (const float*)d_in[11];
  const float* atts_p  = (const float*)d_in[12];
  const float* attd_p  = (const float*)d_in[13];
  const float* Wres_p  = (const float*)d_in[14];
  const float* bias_p  = (const float*)d_in[15];
  const float* lnu_g   = (const float*)d_in[16];
  const float* lnu_b   = (const float*)d_in[17];
  const float* Wu      = (const float*)d_in[18];
  const float* atts_u  = (const float*)d_in[19];
  const float* attd_u  = (const float*)d_in[20];
  const float* Wres_u  = (const float*)d_in[21];
  const float* bias_u  = (const float*)d_in[22];

  const int* esrc = ei;
  const int* edst = ei + NE;

  float* ws     = (float*)d_ws;
  float* p_proj = ws;                                   // N*D
  float* upd_u  = p_proj + (size_t)NN * DD;             // N*D
  float* xh     = upd_u + (size_t)NN * DD;              // N*D
  float* mu     = xh + (size_t)NN * DD;                 // N
  float* rstd   = mu + NN;                              // N
  float* a_src  = rstd + NN;                            // N*H
  float* a_dst  = a_src + (size_t)NN * HH;              // N*H
  unsigned* amax = (unsigned*)(a_dst + (size_t)NN * HH);// N*H
  float* denom  = (float*)(amax + (size_t)NN * HH);     // N*H

  const int ROWSTAT_BLOCKS = NN / 8;                    // 1 wave/row, 8/block
  const int GEMM_BLOCKS    = (NN / 16) * (DD / 16) / 8; // 1 wave/tile, 8/block
  const int NH_BLOCKS      = (NN * HH + 255) / 256;
  const int ETHR_BLOCKS    = (TOTE + 255) / 256;
  const int EWAVE_BLOCKS   = (TOTE + 7) / 8;            // 1 wave/edge

  // ---- front: p_proj = LN(cat(ps,pf,pb)) @ fc_W.T + fc_b ----
  rowstat3_kernel<<<ROWSTAT_BLOCKS, 256, 0, stream>>>(ps, pf, pb, mu, rstd);
  gemm_cat_wmma<<<GEMM_BLOCKS, 256, 0, stream>>>(ps, pf, pb, mu, rstd, ln0_g,
                                                 ln0_b, fc_W, fc_b, p_proj);

  // ---- generic GAT layer ----
  auto gat = [&](const float* X, const float* g, const float* b,
                 const float* W, const float* as, const float* ad,
                 const float* Wres, const float* bias, float* outbuf) {
    rowstat1_kernel<<<ROWSTAT_BLOCKS, 256, 0, stream>>>(X, mu, rstd);
    gemm_ln_wmma<<<GEMM_BLOCKS, 256, 0, stream>>>(X, mu, rstd, g, b, W,
                                                  nullptr, xh);
    gemm_ln_wmma<<<GEMM_BLOCKS, 256, 0, stream>>>(X, mu, rstd, g, b, Wres,
                                                  bias, outbuf);
    attn_kernel<<<NH_BLOCKS, 256, 0, stream>>>(xh, as, ad, a_src, a_dst);
    init_kernel<<<NH_BLOCKS, 256, 0, stream>>>(amax, denom);
    edge_max_kernel<<<ETHR_BLOCKS, 256, 0, stream>>>(esrc, edst, a_src, a_dst,
                                                     amax);
    edge_sum_kernel<<<ETHR_BLOCKS, 256, 0, stream>>>(esrc, edst, a_src, a_dst,
                                                     amax, denom);
    edge_aggr_kernel<<<EWAVE_BLOCKS, 256, 0, stream>>>(esrc, edst, a_src,
                                                       a_dst, amax, denom, xh,
                                                       outbuf);
  };

  // layer 1: upd_u = GAT(LN(p_proj))
  gat(p_proj, lnp_g, lnp_b, Wp, atts_p, attd_p, Wres_p, bias_p, upd_u);
  // layer 2: out = GAT(LN(upd_u))
  gat(upd_u, lnu_g, lnu_b, Wu, atts_u, attd_u, Wres_u, bias_u, (float*)d_out);
}